// TransformerLikeMVBlock_77120432767258
// MI455X (gfx1250) — compile-verified
//
#include <hip/hip_runtime.h>

// ============================================================================
// Geometric-algebra transformer block (Cl(3,0)) for gfx1250 / MI455X.
// All contractions mapped to V_WMMA_F32_16X16X4_F32 (fp32 matrix pipe).
// One wave32 processes a tile of 16 batch multivectors.
// ============================================================================

typedef float v2f __attribute__((ext_vector_type(2)));
typedef float v8f __attribute__((ext_vector_type(8)));

#define WPB 2            // waves per block
#define NT  (WPB * 32)   // threads per block

// ---- Cl(3,0) compile-time tables (blade order: 1, e1,e2,e3, e12,e13,e23, e123)
constexpr int h_MASK[8]  = {0, 1, 2, 4, 3, 5, 6, 7};  // blade idx -> bit mask
constexpr int h_IDX[8]   = {0, 1, 2, 4, 3, 5, 6, 7};  // bit mask -> blade idx
constexpr int h_GRADE[8] = {0, 1, 1, 1, 2, 2, 2, 3};

constexpr int h_popc(int x) { int c = 0; while (x) { c += x & 1; x >>= 1; } return c; }
constexpr int h_sign(int a, int b) {
  int c = 0; int t = a >> 1;
  while (t) { c += h_popc(t & b); t >>= 1; }
  return (c & 1) ? -1 : 1;
}

// PATH_IDX = argwhere(PATHS) in C order; 20 grade-product paths of Cl(3,0)
constexpr int h_PIDX[4][4][4] = {
  { { 0,-1,-1,-1}, {-1, 1,-1,-1}, {-1,-1, 2,-1}, {-1,-1,-1, 3} },
  { {-1, 4,-1,-1}, { 5,-1, 6,-1}, {-1, 7,-1, 8}, {-1,-1, 9,-1} },
  { {-1,-1,10,-1}, {-1,11,-1,12}, {13,-1,14,-1}, {-1,15,-1,-1} },
  { {-1,-1,-1,16}, {-1,-1,17,-1}, {-1,18,-1,-1}, {19,-1,-1,-1} },
};

struct GPTab {
  int   kk[8][8];   // xr blade index k with mask(i)^mask(k)=mask(j)
  float sg[8][8];   // Cayley sign for (i, k(i,j))
  int   pp[8][8];   // w_gp path index for (grade(i), grade(j), grade(k))
};
constexpr GPTab h_mk() {
  GPTab t{};
  for (int i = 0; i < 8; ++i)
    for (int j = 0; j < 8; ++j) {
      int mk = h_MASK[i] ^ h_MASK[j];
      int k  = h_IDX[mk];
      t.kk[i][j] = k;
      t.sg[i][j] = (float)h_sign(h_MASK[i], mk);
      t.pp[i][j] = h_PIDX[h_GRADE[i]][h_GRADE[j]][h_GRADE[k]];
    }
  return t;
}
constexpr GPTab TAB = h_mk();
__device__ constexpr int GR[8] = {0, 1, 1, 1, 2, 2, 2, 3};

__device__ __forceinline__ v8f wmma4(v2f a, v2f b, v8f c) {
  // D(16x16,f32) = A(16x4,f32) * B(4x16,f32) + C
  return __builtin_amdgcn_wmma_f32_16x16x4_f32(false, a, false, b, (short)0, c,
                                               false, false);
}

__device__ __forceinline__ float dsig(float v) { return 1.0f / (1.0f + __expf(-v)); }

// Hard-cap at 256 VGPRs: keeps all operands in the directly-addressable bank,
// eliminating s_set_vgpr_msb mode churn and cross-bank v_mov copies.
__global__ __launch_bounds__(NT)
__attribute__((amdgpu_num_vgpr(256)))
void ga_block_kernel(
    const float* __restrict__ x,       const float* __restrict__ w_right,
    const float* __restrict__ a_norm,  const float* __restrict__ w_gp,
    const float* __restrict__ w_left,  const float* __restrict__ b_left,
    const float* __restrict__ w_up,    const float* __restrict__ b_up,
    const float* __restrict__ a_act,   const float* __restrict__ b_act,
    const float* __restrict__ w_down,  const float* __restrict__ b_down,
    float* __restrict__ out, int nTiles)
{
  __shared__ float s_wgp[20][16][16];          // [path][n][m]
  __shared__ float s_wr [4][16][16];           // [grade][n][m]
  __shared__ float s_wl [4][16][16];
  __shared__ float s_bl[16], s_bu[64], s_bd[16];
  __shared__ float s_bufA[WPB][16][8][17];     // [wave][ch][blade][batch] (pad 17)
  __shared__ float s_bufB[WPB][16][8][17];

  const int tid = threadIdx.x;

  // ---- stage small weights once per block (transposed for conflict-free B-frags)
  for (int t = tid; t < 20 * 256; t += NT) {
    int p = t >> 8, n = (t >> 4) & 15, m = t & 15;
    s_wgp[p][n][m] = w_gp[(m * 16 + n) * 20 + p];
  }
  for (int t = tid; t < 4 * 256; t += NT) {
    int g = t >> 8, n = (t >> 4) & 15, m = t & 15;
    s_wr[g][n][m] = w_right[(m * 16 + n) * 4 + g];
    s_wl[g][n][m] = w_left [(m * 16 + n) * 4 + g];
  }
  if (tid < 16) { s_bl[tid] = b_left[tid]; s_bd[tid] = b_down[tid]; }
  if (tid < 64) { s_bu[tid] = b_up[tid]; }
  __syncthreads();

  const int wave = tid >> 5;
  const int lane = tid & 31;
  const int half = lane >> 4;        // which K-pair this lane carries in frags
  const int rr   = lane & 15;        // A: row M / batch; B,C,D: column N
  const int kb   = half << 1;        // K base within a K=4 chunk

  float (*bufA)[8][17] = s_bufA[wave];
  float (*bufB)[8][17] = s_bufB[wave];

  // tile-invariant: sigmoid of norm gates (channel = rr)
  float sgn[4];
#pragma unroll
  for (int g = 0; g < 4; ++g) sgn[g] = dsig(a_norm[rr * 4 + g]);

  const int waveGlobal = blockIdx.x * WPB + wave;
  const int waveStride = gridDim.x * WPB;

  for (int tile = waveGlobal; tile < nTiles; tile += waveStride) {
    const float* xg = x + (size_t)tile * (16 * 128);

    // ---- load x tile -> bufA[ch][blade][b]
#pragma unroll
    for (int k = 0; k < 16; ++k) {
      int flat = (lane + k * 32) * 4;
      float4 v = *(const float4*)(xg + flat);
      int b = flat >> 7, n = (flat >> 3) & 15, i0 = flat & 7;
      bufA[n][i0 + 0][b] = v.x;
      bufA[n][i0 + 1][b] = v.y;
      bufA[n][i0 + 2][b] = v.z;
      bufA[n][i0 + 3][b] = v.w;
    }
    __builtin_amdgcn_wave_barrier();

    // ---- right linear: xr[blade] = x @ w_right[grade]   (8 x 16x16x16)
    v8f xr[8];
#pragma unroll
    for (int i = 0; i < 8; ++i) {
      v8f acc = {0.f, 0.f, 0.f, 0.f, 0.f, 0.f, 0.f, 0.f};
#pragma unroll
      for (int c = 0; c < 4; ++c) {
        int K0 = 4 * c + kb;
        v2f a, b;
        a.x = bufA[K0][i][rr];      a.y = bufA[K0 + 1][i][rr];
        b.x = s_wr[GR[i]][K0][rr];  b.y = s_wr[GR[i]][K0 + 1][rr];
        acc = wmma4(a, b, acc);
      }
      xr[i] = acc;
    }

    // ---- gated per-grade normalization (elementwise in C/D layout)
#pragma unroll
    for (int r = 0; r < 8; ++r) {
      float v0 = xr[0][r], v1 = xr[1][r], v2 = xr[2][r], v3 = xr[3][r];
      float v4 = xr[4][r], v5 = xr[5][r], v6 = xr[6][r], v7 = xr[7][r];
      float n0 = sqrtf(v0 * v0);
      float n1 = sqrtf(v1 * v1 + v2 * v2 + v3 * v3);
      float n2 = sqrtf(v4 * v4 + v5 * v5 + v6 * v6);
      float n3 = sqrtf(v7 * v7);
      float f0 = 1.f / (sgn[0] * (n0 - 1.f) + 1.f + 1e-6f);
      float f1 = 1.f / (sgn[1] * (n1 - 1.f) + 1.f + 1e-6f);
      float f2 = 1.f / (sgn[2] * (n2 - 1.f) + 1.f + 1e-6f);
      float f3 = 1.f / (sgn[3] * (n3 - 1.f) + 1.f + 1e-6f);
      xr[0][r] = v0 * f0; xr[1][r] = v1 * f1; xr[2][r] = v2 * f1; xr[3][r] = v3 * f1;
      xr[4][r] = v4 * f2; xr[5][r] = v5 * f2; xr[6][r] = v6 * f2; xr[7][r] = v7 * f3;
    }
    // xr -> bufB[ch][blade][b]
#pragma unroll
    for (int i = 0; i < 8; ++i)
#pragma unroll
      for (int r = 0; r < 8; ++r)
        bufB[rr][i][r + 8 * half] = xr[i][r];
    __builtin_amdgcn_wave_barrier();

    // ---- attended = (x@w_left + b_left + geometric_product) / sqrt(2)
    v8f att[8];
#pragma unroll
    for (int j = 0; j < 8; ++j) {
      v8f acc = {0.f, 0.f, 0.f, 0.f, 0.f, 0.f, 0.f, 0.f};
#pragma unroll
      for (int c = 0; c < 4; ++c) {
        int K0 = 4 * c + kb;
        v2f a, b;
        a.x = bufA[K0][j][rr];      a.y = bufA[K0 + 1][j][rr];
        b.x = s_wl[GR[j]][K0][rr];  b.y = s_wl[GR[j]][K0 + 1][rr];
        acc = wmma4(a, b, acc);
      }
      att[j] = acc;
    }
    // geometric product: per output blade j, K=128 over (n,i):
    //   A[b,K] = sign(i,j) * x[b,n,i] * xr[b,n,i^j],  B[K,m] = w_gp[m,n,path(i,j)]
#pragma unroll 1
    for (int n = 0; n < 16; ++n) {
#pragma unroll
      for (int ql = 0; ql < 2; ++ql) {
        const int i00 = ql * 4;
        const int iA = i00 + kb;                // lane-dependent i
        float xv0 = bufA[n][iA][rr];
        float xv1 = bufA[n][iA + 1][rr];
#pragma unroll
        for (int j = 0; j < 8; ++j) {
          // compile-time signs (post-unroll constants): ±1 multiplies fold to
          // neg source-modifiers; equal-arm selects collapse to one value.
          const float slo0 = TAB.sg[i00 + 0][j], shi0 = TAB.sg[i00 + 2][j];
          const float slo1 = TAB.sg[i00 + 1][j], shi1 = TAB.sg[i00 + 3][j];
          float xs0 = half ? (shi0 * xv0) : (slo0 * xv0);
          float xs1 = half ? (shi1 * xv1) : (slo1 * xv1);
          int k0 = half ? TAB.kk[i00 + 2][j] : TAB.kk[i00 + 0][j];
          int k1 = half ? TAB.kk[i00 + 3][j] : TAB.kk[i00 + 1][j];
          int p0 = half ? TAB.pp[i00 + 2][j] : TAB.pp[i00 + 0][j];
          int p1 = half ? TAB.pp[i00 + 3][j] : TAB.pp[i00 + 1][j];
          v2f a, b;
          a.x = xs0 * bufB[n][k0][rr];
          a.y = xs1 * bufB[n][k1][rr];
          b.x = s_wgp[p0][n][rr];
          b.y = s_wgp[p1][n][rr];
          att[j] = wmma4(a, b, att[j]);
        }
      }
    }
    {
      float bl = s_bl[rr];
#pragma unroll
      for (int r = 0; r < 8; ++r) att[0][r] += bl;
#pragma unroll
      for (int j = 0; j < 8; ++j)
#pragma unroll
        for (int r = 0; r < 8; ++r) att[j][r] *= 0.70710678118654752f;
    }

    // attended -> bufA (x is dead), keep as residual accumulator in regs
    __builtin_amdgcn_wave_barrier();
#pragma unroll
    for (int j = 0; j < 8; ++j)
#pragma unroll
      for (int r = 0; r < 8; ++r)
        bufA[rr][j][r + 8 * half] = att[j][r];
    __builtin_amdgcn_wave_barrier();

    v8f outv[8];
#pragma unroll
    for (int j = 0; j < 8; ++j) outv[j] = att[j];

    // ---- MLP: 4 hidden chunks of 16 channels
#pragma unroll 1
    for (int hc = 0; hc < 4; ++hc) {
      const int mh = hc * 16 + rr;             // hidden channel for this lane
      v8f h[8];
#pragma unroll
      for (int j = 0; j < 8; ++j) {
        v8f acc = {0.f, 0.f, 0.f, 0.f, 0.f, 0.f, 0.f, 0.f};
#pragma unroll
        for (int c = 0; c < 4; ++c) {
          int K0 = 4 * c + kb;
          v2f a, b;
          a.x = bufA[K0][j][rr];  a.y = bufA[K0 + 1][j][rr];
          b.x = w_up[(mh * 16 + K0) * 4 + GR[j]];       // from L2
          b.y = w_up[(mh * 16 + K0 + 1) * 4 + GR[j]];
          acc = wmma4(a, b, acc);
        }
        h[j] = acc;
      }
      // bias + multivector SiLU gating
      float bu = s_bu[mh];
      float aa[4], bb[4];
#pragma unroll
      for (int g = 0; g < 4; ++g) { aa[g] = a_act[mh * 4 + g]; bb[g] = b_act[mh * 4 + g]; }
#pragma unroll
      for (int r = 0; r < 8; ++r) {
        float h0 = h[0][r] + bu;
        float h1 = h[1][r], h2 = h[2][r], h3 = h[3][r];
        float h4 = h[4][r], h5 = h[5][r], h6 = h[6][r], h7 = h[7][r];
        float i1 = h1 * h1 + h2 * h2 + h3 * h3;
        float i2 = h4 * h4 + h5 * h5 + h6 * h6;
        float i3 = h7 * h7;
        float g0 = dsig(aa[0] * h0 + bb[0]);
        float g1 = dsig(aa[1] * i1 + bb[1]);
        float g2 = dsig(aa[2] * i2 + bb[2]);
        float g3 = dsig(aa[3] * i3 + bb[3]);
        h[0][r] = h0 * g0; h[1][r] = h1 * g1; h[2][r] = h2 * g1; h[3][r] = h3 * g1;
        h[4][r] = h4 * g2; h[5][r] = h5 * g2; h[6][r] = h6 * g2; h[7][r] = h7 * g3;
      }
      // h -> bufB (xr is dead)
      __builtin_amdgcn_wave_barrier();
#pragma unroll
      for (int j = 0; j < 8; ++j)
#pragma unroll
        for (int r = 0; r < 8; ++r)
          bufB[rr][j][r + 8 * half] = h[j][r];
      __builtin_amdgcn_wave_barrier();
      // down projection accumulates into residual
#pragma unroll
      for (int j = 0; j < 8; ++j) {
#pragma unroll
        for (int c = 0; c < 4; ++c) {
          int K0 = 4 * c + kb;
          v2f a, b;
          a.x = bufB[K0][j][rr];  a.y = bufB[K0 + 1][j][rr];
          b.x = w_down[(rr * 64 + hc * 16 + K0) * 4 + GR[j]];
          b.y = w_down[(rr * 64 + hc * 16 + K0 + 1) * 4 + GR[j]];
          outv[j] = wmma4(a, b, outv[j]);
        }
      }
      __builtin_amdgcn_wave_barrier();
    }
    {
      float bd = s_bd[rr];
#pragma unroll
      for (int r = 0; r < 8; ++r) outv[0][r] += bd;
    }

    // ---- store: out[b, ch, blade]; blades contiguous -> 2x b128 per row
    float* og = out + (size_t)tile * (16 * 128);
#pragma unroll
    for (int r = 0; r < 8; ++r) {
      int brow = r + 8 * half;
      float4 lo = make_float4(outv[0][r], outv[1][r], outv[2][r], outv[3][r]);
      float4 hi = make_float4(outv[4][r], outv[5][r], outv[6][r], outv[7][r]);
      *(float4*)(og + brow * 128 + rr * 8 + 0) = lo;
      *(float4*)(og + brow * 128 + rr * 8 + 4) = hi;
    }
    __builtin_amdgcn_wave_barrier();   // before next tile overwrites bufA
  }
}

extern "C" void kernel_launch(void* const* d_in, const int* in_sizes, int n_in,
                              void* d_out, int out_size, void* d_ws, size_t ws_size,
                              hipStream_t stream) {
  (void)n_in; (void)out_size; (void)d_ws; (void)ws_size;
  const float* x       = (const float*)d_in[0];
  const float* w_right = (const float*)d_in[1];
  const float* a_norm  = (const float*)d_in[2];
  const float* w_gp    = (const float*)d_in[3];
  const float* w_left  = (const float*)d_in[4];
  const float* b_left  = (const float*)d_in[5];
  const float* w_up    = (const float*)d_in[6];
  const float* b_up    = (const float*)d_in[7];
  const float* a_act   = (const float*)d_in[8];
  const float* b_act   = (const float*)d_in[9];
  const float* w_down  = (const float*)d_in[10];
  const float* b_down  = (const float*)d_in[11];

  int Bsz    = in_sizes[0] / 128;   // [B,16,8]
  int nTiles = Bsz / 16;
  if (nTiles <= 0) return;
  int blocks = (nTiles + WPB - 1) / WPB;
  if (blocks > 2048) blocks = 2048;

  ga_block_kernel<<<blocks, NT, 0, stream>>>(
      x, w_right, a_norm, w_gp, w_left, b_left, w_up, b_up, a_act, b_act,
      w_down, b_down, (float*)d_out, nTiles);
}